// GeometricScramblingDiagnostic_31181462569458
// MI455X (gfx1250) — compile-verified
//
#include <hip/hip_runtime.h>
#include <math.h>

// MI455X / gfx1250, wave32. Persistent per-wave on-chip leapfrog.
// q/p state lives in registers in WMMA-A-fragment layout (q regs feed
// v_wmma_f32_16x16x4_f32 directly); weights + activation transposes in LDS;
// all row reductions via wave shuffles; all divides via v_rcp_f32.

typedef __attribute__((ext_vector_type(2))) float v2f;
typedef __attribute__((ext_vector_type(8))) float v8f;

#define LATENT   64
#define HIDDEN   128
#define BPW      8      // batch indices per wave (16-row tile: 8 base + 8 pert)
#define STEPS    100
#define BATCH    2048
#define DT       0.05f
#define DAMPING  0.01f
#define PERT     1e-4f
#define EPSF     1e-8f

__device__ __forceinline__ float frcp(float x)  { return __builtin_amdgcn_rcpf(x); }
__device__ __forceinline__ float fsqrt(float x) { return __builtin_amdgcn_sqrtf(x); }
__device__ __forceinline__ float fsig(float z)  { return frcp(1.0f + __expf(-z)); }
// tanh(x) = 1 - 2/(exp(2x)+1)
__device__ __forceinline__ float ftanh(float x) { return 1.0f - 2.0f * frcp(1.0f + __expf(2.0f * x)); }

__global__ __launch_bounds__(32)
void geo_scramble_kernel(const float* __restrict__ IS, const float* __restrict__ PD,
                         const float* __restrict__ W1, const float* __restrict__ b1,
                         const float* __restrict__ W2, const float* __restrict__ b2,
                         const float* __restrict__ W3, const float* __restrict__ b3,
                         float* __restrict__ out) {
  // ---- LDS (~97.5 KB; 3 single-wave workgroups fit per 320 KB WGP) ----
  __shared__ float sW1[LATENT][HIDDEN];   // 32 KB
  __shared__ float sW2[HIDDEN][LATENT];   // 32 KB
  __shared__ float sb1[HIDDEN];
  __shared__ float sb2[LATENT];
  __shared__ float sW3[LATENT];
  __shared__ float sH1[16][HIDDEN];       // silu(z1)   (GEMM1 -> GEMM2 transpose)
  __shared__ float sD1[16][HIDDEN];       // silu'(z1)
  __shared__ float sG2[16][LATENT];       // silu'(z2)*W3
  __shared__ float sG1[16][HIDDEN];       // gated backprop
  __shared__ float sGrad[16][LATENT];     // dV/dq (GEMM4 D -> elementwise transpose)

  const int lane = threadIdx.x;           // wave32
  const int n_   = lane & 15;             // D column / B column / A row
  const int kk   = lane >> 4;             // lane-group (0,1)
  const int row  = n_;                    // owned tile row (0-7 base, 8-15 pert)
  const int b0   = blockIdx.x * BPW;
  const int bi   = b0 + (row & 7);

  // ---- weights -> LDS (vectorized b128) ----
  {
    float4*       w1d = (float4*)&sW1[0][0];
    const float4* w1s = (const float4*)W1;
    for (int i = lane; i < LATENT * HIDDEN / 4; i += 32) w1d[i] = w1s[i];
    float4*       w2d = (float4*)&sW2[0][0];
    const float4* w2s = (const float4*)W2;
    for (int i = lane; i < HIDDEN * LATENT / 4; i += 32) w2d[i] = w2s[i];
    for (int i = lane; i < HIDDEN; i += 32) sb1[i] = b1[i];
    for (int i = lane; i < LATENT; i += 32) { sb2[i] = b2[i]; sW3[i] = W3[i]; }
    (void)b3; // potential value never needed, only its gradient
  }

  // ---- register state in WMMA A-fragment layout ----
  // qf[k] holds q[row][4k + 2*kk + {0,1}]  ==  A-fragment for k-th K-step.
  v2f qf[16], pf[16], gf[16];
  #pragma unroll
  for (int k = 0; k < 16; ++k) {
    const int c = 4 * k + 2 * kk;
    float x0 = IS[bi * LATENT + c];
    float x1 = IS[bi * LATENT + c + 1];
    if (row >= 8) {
      x0 += PERT * PD[bi * LATENT + c];
      x1 += PERT * PD[bi * LATENT + c + 1];
    }
    qf[k].x = x0; qf[k].y = x1;
    pf[k].x = 0.0f; pf[k].y = 0.0f;
  }

  // row-sum across the two half-lanes (lane ^ 16 owns the complementary columns)
  auto xsum16 = [&](float p) -> float { return p + __shfl_xor(p, 16, 32); };

  // clamp perturbed rows to the ball
  {
    float s = 0.0f;
    #pragma unroll
    for (int k = 0; k < 16; ++k) s += qf[k].x * qf[k].x + qf[k].y * qf[k].y;
    const float qn = fsqrt(xsum16(s));
    if (row >= 8 && qn > 0.99f) {
      const float f = 0.99f * frcp(qn + EPSF);
      #pragma unroll
      for (int k = 0; k < 16; ++k) { qf[k].x *= f; qf[k].y *= f; }
    }
  }
  __syncthreads();

  // ---- gradV: 4 fp32 WMMA GEMMs; A of GEMM1 comes straight from qf regs ----
  auto gradV = [&]() {
    // GEMM1: z1[16,128] = Q @ W1 + b1 -> H1 = silu, D1 = silu'
    for (int j = 0; j < 8; ++j) {
      v8f acc;
      const float bb = sb1[j * 16 + n_];
      #pragma unroll
      for (int v = 0; v < 8; ++v) acc[v] = bb;
      #pragma unroll
      for (int k = 0; k < 16; ++k) {
        v2f b;
        b.x = sW1[4 * k + 2 * kk    ][j * 16 + n_];
        b.y = sW1[4 * k + 2 * kk + 1][j * 16 + n_];
        acc = __builtin_amdgcn_wmma_f32_16x16x4_f32(false, qf[k], false, b, (short)0, acc, false, false);
      }
      #pragma unroll
      for (int v = 0; v < 8; ++v) {
        const int r = v + 8 * kk, c = j * 16 + n_;
        const float z = acc[v];
        const float s = fsig(z);
        sH1[r][c] = z * s;
        sD1[r][c] = s * (1.0f + z * (1.0f - s));
      }
    }
    __syncthreads();
    // GEMM2: z2[16,64] = H1 @ W2 + b2 -> G2 = silu'(z2) * W3[col]
    for (int j = 0; j < 4; ++j) {
      v8f acc;
      const float bb = sb2[j * 16 + n_];
      #pragma unroll
      for (int v = 0; v < 8; ++v) acc[v] = bb;
      #pragma unroll
      for (int k = 0; k < 32; ++k) {
        v2f a, b;
        a.x = sH1[n_][4 * k + 2 * kk];
        a.y = sH1[n_][4 * k + 2 * kk + 1];
        b.x = sW2[4 * k + 2 * kk    ][j * 16 + n_];
        b.y = sW2[4 * k + 2 * kk + 1][j * 16 + n_];
        acc = __builtin_amdgcn_wmma_f32_16x16x4_f32(false, a, false, b, (short)0, acc, false, false);
      }
      const float w3c = sW3[j * 16 + n_];
      #pragma unroll
      for (int v = 0; v < 8; ++v) {
        const int r = v + 8 * kk, c = j * 16 + n_;
        const float z = acc[v];
        const float s = fsig(z);
        sG2[r][c] = s * (1.0f + z * (1.0f - s)) * w3c;
      }
    }
    __syncthreads();
    // GEMM3: G1[16,128] = (G2 @ W2^T) .* D1
    for (int j = 0; j < 8; ++j) {
      v8f acc;
      #pragma unroll
      for (int v = 0; v < 8; ++v) acc[v] = 0.0f;
      #pragma unroll
      for (int k = 0; k < 16; ++k) {
        v2f a, b;
        a.x = sG2[n_][4 * k + 2 * kk];
        a.y = sG2[n_][4 * k + 2 * kk + 1];
        b.x = sW2[j * 16 + n_][4 * k + 2 * kk];      // W2^T[k'][n] = W2[n][k']
        b.y = sW2[j * 16 + n_][4 * k + 2 * kk + 1];
        acc = __builtin_amdgcn_wmma_f32_16x16x4_f32(false, a, false, b, (short)0, acc, false, false);
      }
      #pragma unroll
      for (int v = 0; v < 8; ++v) {
        const int r = v + 8 * kk, c = j * 16 + n_;
        sG1[r][c] = acc[v] * sD1[r][c];
      }
    }
    __syncthreads();
    // GEMM4: grad[16,64] = G1 @ W1^T
    for (int j = 0; j < 4; ++j) {
      v8f acc;
      #pragma unroll
      for (int v = 0; v < 8; ++v) acc[v] = 0.0f;
      #pragma unroll
      for (int k = 0; k < 32; ++k) {
        v2f a, b;
        a.x = sG1[n_][4 * k + 2 * kk];
        a.y = sG1[n_][4 * k + 2 * kk + 1];
        b.x = sW1[j * 16 + n_][4 * k + 2 * kk];      // W1^T[k'][n] = W1[n][k']
        b.y = sW1[j * 16 + n_][4 * k + 2 * kk + 1];
        acc = __builtin_amdgcn_wmma_f32_16x16x4_f32(false, a, false, b, (short)0, acc, false, false);
      }
      #pragma unroll
      for (int v = 0; v < 8; ++v) {
        const int r = v + 8 * kk, c = j * 16 + n_;
        sGrad[r][c] = acc[v];
      }
    }
    __syncthreads();
  };

  // grad -> registers in the same interleaved elementwise layout
  auto loadGrad = [&]() {
    #pragma unroll
    for (int k = 0; k < 16; ++k)
      gf[k] = *(const v2f*)&sGrad[row][4 * k + 2 * kk];
  };

  gradV();      // grad at q0 (cached: gradV(q_new) == next step's gradV(q))
  loadGrad();

  for (int step = 0; step < STEPS; ++step) {
    // --- hyperbolic distance(base row r, pert row r+8), all from regs/shuffles ---
    {
      float d2p = 0.0f, o2p = 0.0f;
      #pragma unroll
      for (int k = 0; k < 16; ++k) {
        const float ox = __shfl_xor(qf[k].x, 8, 32);   // same columns, paired row
        const float oy = __shfl_xor(qf[k].y, 8, 32);
        const float dx = qf[k].x - ox, dy = qf[k].y - oy;
        d2p += dx * dx + dy * dy;
        o2p += qf[k].x * qf[k].x + qf[k].y * qf[k].y;
      }
      const float d2   = xsum16(d2p);
      const float own2 = xsum16(o2p);
      const float oth2 = __shfl_xor(own2, 8, 32);
      if (lane < 8) {                                   // base rows, half 0
        const float x2 = fminf(own2, 0.999f);
        const float y2 = fminf(oth2, 0.999f);
        float arg = 1.0f + 2.0f * d2 * frcp((1.0f - x2) * (1.0f - y2) + EPSF);
        arg = fmaxf(arg, 1.0f + EPSF);
        out[step * BATCH + b0 + lane] = __logf(arg + fsqrt(arg * arg - 1.0f));
      }
    }

    // --- p_half = (p - 0.5*DT*grad) * (1-DAMPING); |q|^2 ---
    float q2p = 0.0f;
    #pragma unroll
    for (int k = 0; k < 16; ++k) {
      pf[k].x = (pf[k].x - 0.5f * DT * gf[k].x) * (1.0f - DAMPING);
      pf[k].y = (pf[k].y - 0.5f * DT * gf[k].y) * (1.0f - DAMPING);
      q2p += qf[k].x * qf[k].x + qf[k].y * qf[k].y;
    }
    const float q2raw = xsum16(q2p);
    const float q2c   = fminf(q2raw, 0.999f);
    const float lam   = 2.0f * frcp(1.0f - q2c);
    const float vscl  = DT * frcp(lam * lam);

    // --- v = vscl * p_half ; |v| ; mag = tanh(min(lam*|v|/2, 15)) ---
    v2f vf[16];
    float vn2p = 0.0f;
    #pragma unroll
    for (int k = 0; k < 16; ++k) {
      vf[k].x = vscl * pf[k].x;
      vf[k].y = vscl * pf[k].y;
      vn2p += vf[k].x * vf[k].x + vf[k].y * vf[k].y;
    }
    const float vn   = fsqrt(xsum16(vn2p)) + EPSF;
    const float mag  = ftanh(fminf(lam * vn * 0.5f, 15.0f));
    const float mrcp = mag * frcp(vn);

    // --- u = mag*v/|v| ; xy = q.u ; y2 = u.u ---
    float xyp = 0.0f, y2p = 0.0f;
    #pragma unroll
    for (int k = 0; k < 16; ++k) {
      vf[k].x *= mrcp; vf[k].y *= mrcp;            // now u
      xyp += qf[k].x * vf[k].x + qf[k].y * vf[k].y;
      y2p += vf[k].x * vf[k].x + vf[k].y * vf[k].y;
    }
    const float xy = xsum16(xyp);
    const float y2 = xsum16(y2p);

    // --- Mobius add (unclamped x2, per reference) + ball clamp ---
    const float ca = 1.0f + 2.0f * xy + y2;
    const float cb = 1.0f - q2raw;
    const float dr = frcp(1.0f + 2.0f * xy + q2raw * y2 + EPSF);
    float rn2p = 0.0f;
    #pragma unroll
    for (int k = 0; k < 16; ++k) {
      vf[k].x = (ca * qf[k].x + cb * vf[k].x) * dr;  // now r
      vf[k].y = (ca * qf[k].y + cb * vf[k].y) * dr;
      rn2p += vf[k].x * vf[k].x + vf[k].y * vf[k].y;
    }
    const float rn  = fsqrt(xsum16(rn2p));
    const float fac = (rn > 0.99f) ? 0.99f * frcp(rn + EPSF) : 1.0f;
    #pragma unroll
    for (int k = 0; k < 16; ++k) { qf[k].x = vf[k].x * fac; qf[k].y = vf[k].y * fac; }

    __syncthreads();
    gradV();      // grad at q_new (A fragments read directly from qf)
    loadGrad();

    // --- p = p_half - 0.5*DT*grad_new ---
    #pragma unroll
    for (int k = 0; k < 16; ++k) {
      pf[k].x -= 0.5f * DT * gf[k].x;
      pf[k].y -= 0.5f * DT * gf[k].y;
    }
  }
}

extern "C" void kernel_launch(void* const* d_in, const int* in_sizes, int n_in,
                              void* d_out, int out_size, void* d_ws, size_t ws_size,
                              hipStream_t stream) {
  const float* IS = (const float*)d_in[0];
  const float* PD = (const float*)d_in[1];
  const float* W1 = (const float*)d_in[2];
  const float* b1 = (const float*)d_in[3];
  const float* W2 = (const float*)d_in[4];
  const float* b2 = (const float*)d_in[5];
  const float* W3 = (const float*)d_in[6];
  const float* b3 = (const float*)d_in[7];
  float* out = (float*)d_out;
  (void)in_sizes; (void)n_in; (void)out_size; (void)d_ws; (void)ws_size;
  hipLaunchKernelGGL(geo_scramble_kernel, dim3(BATCH / BPW), dim3(32), 0, stream,
                     IS, PD, W1, b1, W2, b2, W3, b3, out);
}